// RandTrasl_x_5995774345282
// MI455X (gfx1250) — compile-verified
//
#include <hip/hip_runtime.h>
#include <hip/hip_bf16.h>
#include <stdint.h>

// Problem constants (reference is fixed-shape): x [B, C, T, H, W] fp32.
constexpr int kB = 8;
constexpr int kC = 3;
constexpr int kT = 32;
constexpr int kH = 224;
constexpr int kW = 224;
constexpr int kRows   = 8;            // rows staged per block (same t -> same shift)
constexpr int kChunks = kW / 4;       // 56 x float4 per row
constexpr int kSlots  = kRows * kChunks;  // 448

// Native clang vector type: accepted by __builtin_nontemporal_store.
typedef float v4f __attribute__((ext_vector_type(4)));

struct TxTable { int tx[kT]; };

// CDNA5 async DMA: global -> LDS, 16 bytes per active lane, tracked by ASYNCcnt.
// GVS addressing: mem = SGPR64 base + VGPR32 byte offset; LDS dest offset in VDST VGPR.
__device__ __forceinline__ void async_copy_b128(uint32_t lds_byte_off,
                                                uint32_t glb_byte_off,
                                                const void* base) {
  asm volatile("global_load_async_to_lds_b128 %0, %1, %2"
               :
               : "v"(lds_byte_off), "v"(glb_byte_off), "s"(base)
               : "memory");
}

__device__ __forceinline__ void wait_asynccnt0() {
  asm volatile("s_wait_asynccnt 0" ::: "memory");
}

__global__ __launch_bounds__(256) void
tsm_hshift_kernel(const float* __restrict__ in, float* __restrict__ out, TxTable tab) {
  __shared__ float tile[kRows * kW];          // 7 KB of the 320 KB/WGP pool

  const int tid = threadIdx.x;
  const int img = blockIdx.y;                 // (b*C + c)*T + t
  const int t   = img & (kT - 1);
  const int tx  = tab.tx[t];                  // horizontal shift for this frame
  const int need = kW - tx;                   // # source columns actually consumed

  const size_t base = (size_t)img * (size_t)(kH * kW)
                    + (size_t)blockIdx.x * (size_t)(kRows * kW);
  const float* gin  = in  + base;
  float*       gout = out + base;

  const uint32_t lds_base = (uint32_t)(uintptr_t)(&tile[0]);

  // ---- Stage only the needed input prefix of each row via async DMA ----
  // Fully shifted-out frames (tx >= W) issue zero loads: write-only blocks.
#pragma unroll
  for (int i = 0; i < 2; ++i) {
    const int slot = tid + i * 256;
    if (slot < kSlots) {
      const int r = slot / kChunks;
      const int c = slot - r * kChunks;       // float4 chunk index in row
      if (c * 4 < need) {                     // skip never-read columns (saves ~61% reads)
        const uint32_t off = (uint32_t)((r * kW + c * 4) * sizeof(float));
        async_copy_b128(lds_base + off, off, (const void*)gin);
      }
    }
  }
  wait_asynccnt0();                           // this wave's DMA done
  __syncthreads();                            // visible to all 8 waves

  // ---- Shifted gather from LDS, aligned non-temporal B128 stores ----
#pragma unroll
  for (int i = 0; i < 2; ++i) {
    const int slot = tid + i * 256;
    if (slot < kSlots) {
      const int r  = slot / kChunks;
      const int w0 = (slot - r * kChunks) * 4;
      const int s  = w0 - tx;                 // source column of first lane element
      const float* trow = &tile[r * kW];
      v4f v;
      v.x = (s     >= 0) ? trow[s    ] : 0.0f;
      v.y = (s + 1 >= 0) ? trow[s + 1] : 0.0f;
      v.z = (s + 2 >= 0) ? trow[s + 2] : 0.0f;
      v.w = (s + 3 >= 0) ? trow[s + 3] : 0.0f;
      // Output is streamed once: keep it out of L2's way (input is the reused stream).
      __builtin_nontemporal_store(v, (v4f*)(gout + r * kW + w0));
    }
  }
}

extern "C" void kernel_launch(void* const* d_in, const int* in_sizes, int n_in,
                              void* d_out, int out_size, void* d_ws, size_t ws_size,
                              hipStream_t stream) {
  (void)in_sizes; (void)n_in; (void)out_size; (void)d_ws; (void)ws_size;
  const float* x = (const float*)d_in[0];
  float* out = (float*)d_out;

  // Host-side shift table in double precision: matches np.trunc(0.08*dist*W) exactly.
  TxTable tab;
  for (int t = 0; t < kT; ++t) {
    int d = t - kT / 2;
    if (d < 0) d = -d;
    tab.tx[t] = (int)(0.08 * (double)d * (double)kW);   // trunc toward zero (values >= 0)
  }

  dim3 grid(kH / kRows, kB * kC * kT);   // (28, 768) blocks
  tsm_hshift_kernel<<<grid, 256, 0, stream>>>(x, out, tab);
}